// Corr2_16492674417479
// MI455X (gfx1250) — compile-verified
//
#include <hip/hip_runtime.h>
#include <math.h>

typedef float v2f __attribute__((ext_vector_type(2)));
typedef float v8f __attribute__((ext_vector_type(8)));

#define NDISP 81
#define BN 2
#define TPN 4
#define TFN 4
#define C0 256
#define H0 80
#define W0 128
#define HW0 (H0*W0)

// ---------------------------------------------------------------------------
// 1) per-pixel inverse L2 norm of features0 over channels: rn[b*TP+t][h][w]
// ---------------------------------------------------------------------------
__global__ void norms_kernel(const float* __restrict__ f0, float* __restrict__ rn) {
    int idx = blockIdx.x * blockDim.x + threadIdx.x;      // B*TP*H0*W0
    if (idx >= BN * TPN * HW0) return;
    int pix = idx % HW0;
    int nt  = idx / HW0;
    const float* p = f0 + (size_t)nt * C0 * HW0 + pix;
    float s = 0.f;
    #pragma unroll 4
    for (int c = 0; c < C0; ++c) {
        float v = p[(size_t)c * HW0];
        s += v * v;
    }
    float nrm = fmaxf(sqrtf(s), 1e-12f);
    rn[idx] = 1.0f / nrm;
}

// ---------------------------------------------------------------------------
// 2) pointwise displaced correlation: pc[n, d, h, w]  (n = 6 frame pairs)
//    normalization folded in via rn (so normalized features never stored)
// ---------------------------------------------------------------------------
__global__ void pcorr_kernel(const float* __restrict__ f0, const float* __restrict__ rn,
                             float* __restrict__ pc) {
    int idx = blockIdx.x * blockDim.x + threadIdx.x;      // 6*H0*W0
    if (idx >= 6 * HW0) return;
    int pix = idx % HW0;
    int n   = idx / HW0;
    int b = n / 3, tp = n % 3;
    int h = pix / W0, w = pix % W0;
    const float* x1  = f0 + ((size_t)(b * TPN + tp) * C0) * HW0 + pix;
    const float* x2b = f0 + ((size_t)(b * TPN + tp + 1) * C0) * HW0;

    float acc[NDISP];
    #pragma unroll
    for (int d = 0; d < NDISP; ++d) acc[d] = 0.f;

    for (int c = 0; c < C0; ++c) {
        float x1v = x1[(size_t)c * HW0];
        const float* x2c = x2b + (size_t)c * HW0;
        #pragma unroll
        for (int dy = 0; dy < 9; ++dy) {
            int y2 = h + dy - 4;
            bool vy = (unsigned)y2 < (unsigned)H0;
            #pragma unroll
            for (int dx = 0; dx < 9; ++dx) {
                int x2i = w + dx - 4;
                float v = (vy && (unsigned)x2i < (unsigned)W0) ? x2c[y2 * W0 + x2i] : 0.f;
                acc[dy * 9 + dx] += x1v * v;
            }
        }
    }
    float r1 = rn[(b * TPN + tp) * HW0 + pix];
    const float* rn2 = rn + (b * TPN + tp + 1) * HW0;
    #pragma unroll
    for (int dy = 0; dy < 9; ++dy) {
        int y2 = h + dy - 4;
        bool vy = (unsigned)y2 < (unsigned)H0;
        #pragma unroll
        for (int dx = 0; dx < 9; ++dx) {
            int x2i = w + dx - 4;
            float r2 = (vy && (unsigned)x2i < (unsigned)W0) ? rn2[y2 * W0 + x2i] : 0.f;
            pc[((size_t)n * NDISP + dy * 9 + dx) * HW0 + pix] = acc[dy * 9 + dx] * r1 * r2;
        }
    }
}

// ---------------------------------------------------------------------------
// 3) vertical 9-tap box sum (zero padded)
// ---------------------------------------------------------------------------
__global__ void vbox_kernel(const float* __restrict__ pc, float* __restrict__ tmp) {
    int idx = blockIdx.x * blockDim.x + threadIdx.x;      // 6*81*H0*W0
    if (idx >= 6 * NDISP * HW0) return;
    int pix = idx % HW0;
    int ch  = idx / HW0;
    int h = pix / W0, w = pix % W0;
    const float* src = pc + (size_t)ch * HW0 + w;
    float s = 0.f;
    #pragma unroll
    for (int dy = -4; dy <= 4; ++dy) {
        int y = h + dy;
        if ((unsigned)y < (unsigned)H0) s += src[y * W0];
    }
    tmp[idx] = s;
}

// ---------------------------------------------------------------------------
// 4) horizontal 9-tap box sum fused with softmax-weighted pair mix -> corr0
// ---------------------------------------------------------------------------
__global__ void hboxmix_kernel(const float* __restrict__ tmp, const float* __restrict__ past,
                               float* __restrict__ corr0) {
    int idx = blockIdx.x * blockDim.x + threadIdx.x;      // 2*81*H0*W0
    if (idx >= BN * NDISP * HW0) return;
    int pix = idx % HW0;
    int bd  = idx / HW0;
    int b = bd / NDISP, d = bd % NDISP;
    int h = pix / W0, w = pix % W0;

    float p0 = past[b * TPN + 0], p1 = past[b * TPN + 1], p2 = past[b * TPN + 2];
    float m  = fmaxf(p0, fmaxf(p1, p2));
    float e0 = __expf(p0 - m), e1 = __expf(p1 - m), e2 = __expf(p2 - m);
    float inv = 1.f / (e0 + e1 + e2);
    float wt[3] = {e0 * inv, e1 * inv, e2 * inv};

    float s = 0.f;
    #pragma unroll
    for (int t = 0; t < 3; ++t) {
        const float* src = tmp + ((size_t)((b * 3 + t) * NDISP + d)) * HW0 + h * W0;
        float hs = 0.f;
        #pragma unroll
        for (int dx = -4; dx <= 4; ++dx) {
            int x = w + dx;
            if ((unsigned)x < (unsigned)W0) hs += src[x];
        }
        s += wt[t] * hs;
    }
    corr0[idx] = s;
}

// ---------------------------------------------------------------------------
// 5) bilinear resize (align_corners=True) corr0 -> corr at (Hi, Wi)
// ---------------------------------------------------------------------------
__global__ void resize_kernel(const float* __restrict__ src, float* __restrict__ dst,
                              int Hi, int Wi) {
    int idx = blockIdx.x * blockDim.x + threadIdx.x;      // 2*81*Hi*Wi
    int total = BN * NDISP * Hi * Wi;
    if (idx >= total) return;
    int wi = idx % Wi;
    int r  = idx / Wi;
    int hi = r % Hi;
    int ch = r / Hi;
    float sy = (float)(H0 - 1) / (float)(Hi - 1);
    float sx = (float)(W0 - 1) / (float)(Wi - 1);
    float yf = hi * sy, xf = wi * sx;
    int y0 = (int)floorf(yf); int y1 = y0 + 1; if (y1 > H0 - 1) y1 = H0 - 1;
    int x0 = (int)floorf(xf); int x1 = x0 + 1; if (x1 > W0 - 1) x1 = W0 - 1;
    float wy = yf - (float)y0, wx = xf - (float)x0;
    const float* s = src + (size_t)ch * HW0;
    float v00 = s[y0 * W0 + x0], v01 = s[y0 * W0 + x1];
    float v10 = s[y1 * W0 + x0], v11 = s[y1 * W0 + x1];
    float r0 = v00 * (1.f - wx) + v01 * wx;
    float r1v = v10 * (1.f - wx) + v11 * wx;
    dst[idx] = r0 * (1.f - wy) + r1v * wy;
}

// ---------------------------------------------------------------------------
// 6) fused fp32 WMMA GEMM + affine + TF-expansion + SiLU
//    G1 = W[:, :C]  * f_last   (K = C)
//    G2 = W[:, C:]  * corr     (K = 81, padded to 84)
//    out[b,t,o,p] = SiLU(scale[o]*(G1 + fut[b,t]*G2) + bias[o])
//    One wave -> 16 output channels x 64 pixels; 4 waves/block -> 64x64 tile.
// ---------------------------------------------------------------------------
__launch_bounds__(128)
__global__ void gemm_silu_kernel(const float* __restrict__ f,      // [B,TP,C,H,W]
                                 const float* __restrict__ wmat,   // [C, C+81]
                                 const float* __restrict__ scale,  // [C]
                                 const float* __restrict__ bias,   // [C]
                                 const float* __restrict__ corr,   // [B,81,H,W]
                                 const float* __restrict__ fut,    // [B,TF]
                                 float* __restrict__ out,          // [B,TF,C,H,W]
                                 int C, int HW) {
    const int Ktot = C + NDISP;
    int lane = threadIdx.x & 31;
    int wave = threadIdx.x >> 5;
    int ln = lane & 15;       // column (pixel) / A row within tile
    int lh = lane >> 4;       // K-half selector
    int P0  = blockIdx.x * 64;            // pixel tile base (HW % 64 == 0)
    int b   = P0 / HW;
    int inb = P0 % HW;
    int o0  = blockIdx.y * 64 + wave * 16;

    const float* flast = f + ((size_t)(b * TPN + (TPN - 1)) * C) * HW;
    const float* cb    = corr + (size_t)b * NDISP * HW;
    const float* arow  = wmat + (size_t)(o0 + ln) * Ktot;

    v8f acc1[4], acc2[4];
    #pragma unroll
    for (int pt = 0; pt < 4; ++pt) {
        #pragma unroll
        for (int e = 0; e < 8; ++e) { acc1[pt][e] = 0.f; acc2[pt][e] = 0.f; }
    }

    // ---- K < C : last-frame features (C is a multiple of 4) ----
    for (int k0 = 0; k0 < C; k0 += 4) {
        int ka = k0 + 2 * lh;
        v2f a; a.x = arow[ka]; a.y = arow[ka + 1];
        const float* bsrc = flast + (size_t)ka * HW + inb + ln;
        #pragma unroll
        for (int pt = 0; pt < 4; ++pt) {
            v2f bv; bv.x = bsrc[pt * 16]; bv.y = bsrc[HW + pt * 16];
            acc1[pt] = __builtin_amdgcn_wmma_f32_16x16x4_f32(
                false, a, false, bv, (short)0, acc1[pt], false, false);
        }
    }
    // ---- K in [C, C+81) : correlation channels (pad 81 -> 84 with zeros) ----
    for (int d0 = 0; d0 < 84; d0 += 4) {
        int da = d0 + 2 * lh;
        v2f a;
        a.x = (da     < NDISP) ? arow[C + da]     : 0.f;
        a.y = (da + 1 < NDISP) ? arow[C + da + 1] : 0.f;
        const float* bsrc = cb + (size_t)da * HW + inb + ln;
        #pragma unroll
        for (int pt = 0; pt < 4; ++pt) {
            v2f bv;
            bv.x = (da     < NDISP) ? bsrc[pt * 16]      : 0.f;
            bv.y = (da + 1 < NDISP) ? bsrc[HW + pt * 16] : 0.f;
            acc2[pt] = __builtin_amdgcn_wmma_f32_16x16x4_f32(
                false, a, false, bv, (short)0, acc2[pt], false, false);
        }
    }

    float ftv[TFN];
    #pragma unroll
    for (int t = 0; t < TFN; ++t) ftv[t] = fut[b * TFN + t];

    // D layout: VGPR r -> row (o0 + r + 8*lh), column (ln)
    #pragma unroll
    for (int r = 0; r < 8; ++r) {
        int o = o0 + r + 8 * lh;
        float sc = scale[o], bb = bias[o];
        #pragma unroll
        for (int pt = 0; pt < 4; ++pt) {
            float g1 = acc1[pt][r];
            float g2 = acc2[pt][r];
            size_t pcol = (size_t)inb + pt * 16 + ln;
            #pragma unroll
            for (int t = 0; t < TFN; ++t) {
                float y = sc * (g1 + ftv[t] * g2) + bb;
                float sv = y / (1.f + __expf(-y));
                out[(((size_t)(b * TFN + t) * C + o) * HW) + pcol] = sv;
            }
        }
    }
}

// ---------------------------------------------------------------------------
extern "C" void kernel_launch(void* const* d_in, const int* in_sizes, int n_in,
                              void* d_out, int out_size, void* d_ws, size_t ws_size,
                              hipStream_t stream) {
    const float* f0    = (const float*)d_in[0];
    const float* w0    = (const float*)d_in[1];
    const float* s0    = (const float*)d_in[2];
    const float* b0    = (const float*)d_in[3];
    const float* f1    = (const float*)d_in[4];
    const float* w1    = (const float*)d_in[5];
    const float* s1    = (const float*)d_in[6];
    const float* b1    = (const float*)d_in[7];
    const float* f2    = (const float*)d_in[8];
    const float* w2    = (const float*)d_in[9];
    const float* s2    = (const float*)d_in[10];
    const float* b2    = (const float*)d_in[11];
    const float* past  = (const float*)d_in[12];
    const float* fut   = (const float*)d_in[13];

    // workspace layout (floats)
    const size_t n_rn    = (size_t)BN * TPN * HW0;                 //    81920
    const size_t n_pc    = (size_t)6 * NDISP * HW0;                //  4976640
    const size_t n_tmp   = n_pc;                                   //  4976640
    const size_t n_c0    = (size_t)BN * NDISP * HW0;               //  1658880
    const size_t n_c1    = (size_t)BN * NDISP * 40 * 64;           //   414720
    const size_t n_c2    = (size_t)BN * NDISP * 20 * 32;           //   103680
    const size_t need    = (n_rn + n_pc + n_tmp + n_c0 + n_c1 + n_c2) * sizeof(float);
    if (ws_size < need) return;

    float* rn    = (float*)d_ws;
    float* pc    = rn + n_rn;
    float* tmp   = pc + n_pc;
    float* corr0 = tmp + n_tmp;
    float* corr1 = corr0 + n_c0;
    float* corr2 = corr1 + n_c1;

    float* out0 = (float*)d_out;
    float* out1 = out0 + (size_t)BN * TFN * 256 * 80 * 128;
    float* out2 = out1 + (size_t)BN * TFN * 512 * 40 * 64;

    // 1) norms
    {
        int n = BN * TPN * HW0;
        norms_kernel<<<(n + 255) / 256, 256, 0, stream>>>(f0, rn);
    }
    // 2) pointwise correlation
    {
        int n = 6 * HW0;
        pcorr_kernel<<<(n + 127) / 128, 128, 0, stream>>>(f0, rn, pc);
    }
    // 3) vertical box
    {
        int n = 6 * NDISP * HW0;
        vbox_kernel<<<(n + 255) / 256, 256, 0, stream>>>(pc, tmp);
    }
    // 4) horizontal box + softmax pair mix
    {
        int n = BN * NDISP * HW0;
        hboxmix_kernel<<<(n + 255) / 256, 256, 0, stream>>>(tmp, past, corr0);
    }
    // 5) resizes
    {
        int n1 = BN * NDISP * 40 * 64;
        resize_kernel<<<(n1 + 255) / 256, 256, 0, stream>>>(corr0, corr1, 40, 64);
        int n2 = BN * NDISP * 20 * 32;
        resize_kernel<<<(n2 + 255) / 256, 256, 0, stream>>>(corr0, corr2, 20, 32);
    }
    // 6) fused WMMA GEMM + SiLU per scale
    {
        dim3 blk(128, 1, 1);
        // scale 0: C=256, HW=10240, pixels N = 2*10240 = 20480
        dim3 g0(20480 / 64, 256 / 64, 1);
        gemm_silu_kernel<<<g0, blk, 0, stream>>>(f0, w0, s0, b0, corr0, fut, out0, 256, HW0);
        // scale 1: C=512, HW=2560, N = 5120
        dim3 g1(5120 / 64, 512 / 64, 1);
        gemm_silu_kernel<<<g1, blk, 0, stream>>>(f1, w1, s1, b1, corr1, fut, out1, 512, 40 * 64);
        // scale 2: C=1024, HW=640, N = 1280
        dim3 g2(1280 / 64, 1024 / 64, 1);
        gemm_silu_kernel<<<g2, blk, 0, stream>>>(f2, w2, s2, b2, corr2, fut, out2, 1024, 20 * 32);
    }
}